// MPS_18545668784598
// MI455X (gfx1250) — compile-verified
//
#include <hip/hip_runtime.h>
#include <math.h>

// Problem constants (B, N, D, R) = (1024, 64, 128, 128)
#define B_    1024
#define N_    64
#define D_    128
#define R_    128
#define NMID  62                      // N-2 middle cores
#define CORE_ELEMS (R_*D_*R_)         // 2,097,152 floats per g_mid core
#define LN2 0.6931471805599453f
#define LSTR 132    // L row stride: even (b64-aligned), banks step 4 -> halves disjoint
#define PSTR 288    // K-pair stride for pair-interleaved arrays: 288 % 64 == 32

typedef float v2f __attribute__((ext_vector_type(2)));
typedef float v8f __attribute__((ext_vector_type(8)));

// ---------------------------------------------------------------------------
// Zero the per-step max slots (ws is poisoned 0xAA; atomicMax needs 0 start).
// ---------------------------------------------------------------------------
__global__ void k_init(unsigned int* __restrict__ maxv) {
  if (threadIdx.x < 64) maxv[threadIdx.x] = 0u;
}

// ---------------------------------------------------------------------------
// psi chain: one workgroup (128 threads) per batch element. Fully independent
// chains; per-b exponent-exact rescale with accumulated log. VALU-bound
// (2 GFLOP total), gathers hit L2 (<=128 distinct matrices per step).
// ---------------------------------------------------------------------------
__global__ __launch_bounds__(128)
void k_psi(const int* __restrict__ x, const float* __restrict__ g0,
           const float* __restrict__ gmid, const float* __restrict__ glast,
           float* __restrict__ psilog) {
  __shared__ float pa[R_], pb[R_];
  const int b = blockIdx.x, j = threadIdx.x;
  pa[j] = g0[x[b * N_] * R_ + j];
  float logscale = 0.f;
  __syncthreads();
  for (int step = 0; step < NMID; ++step) {
    const int xi = x[b * N_ + 1 + step];
    const float* G = gmid + (size_t)step * CORE_ELEMS + (size_t)xi * R_ + j;
    float acc = 0.f;
#pragma unroll 8
    for (int i = 0; i < R_; ++i) acc = fmaf(pa[i], G[(size_t)i * (D_ * R_)], acc);
    pb[j] = acc;
    __syncthreads();
    float mx = 0.f;                       // redundant scan, LDS broadcast
    for (int i = 0; i < R_; ++i) mx = fmaxf(mx, fabsf(pb[i]));
    const int e = (mx > 0.f) ? ilogbf(mx) : 0;   // exact power-of-2 rescale
    logscale += (float)e * LN2;
    pa[j] = pb[j] * exp2f((float)(-e));
    __syncthreads();
  }
  const int xl = x[b * N_ + (N_ - 1)];
  pb[j] = pa[j] * glast[j * D_ + xl];     // g_last[(j, xl, 0)]
  __syncthreads();
  if (j == 0) {
    float s = 0.f;
    for (int i = 0; i < R_; ++i) s += pb[i];
    psilog[b] = logf(fabsf(s)) + logscale;   // = log|psi_raw_b|
  }
}

// ---------------------------------------------------------------------------
// L0 = g0^T g0  (128x128, tiny)
// ---------------------------------------------------------------------------
__global__ __launch_bounds__(256)
void k_linit(const float* __restrict__ g0, float* __restrict__ L) {
  const int idx = blockIdx.x * 256 + threadIdx.x;  // 64 blocks -> 16384
  const int q = idx >> 7, s = idx & 127;
  float acc = 0.f;
  for (int d = 0; d < D_; ++d) acc = fmaf(g0[d * R_ + q], g0[d * R_ + s], acc);
  L[idx] = acc;
}

// ---------------------------------------------------------------------------
// One L-chain step: WG d computes P_d = A_d^T (L A_d) with exact-fp32 WMMA.
// Register-blocked (8 accumulator tiles per wave) + LDS layouts chosen so
// every WMMA operand is ONE aligned ds_load_b64:
//   Ls: row-major stride 132 (a-operand (L[m,k0],L[m,k0+1]) contiguous)
//   Ap/Mp: K-pair interleaved, (k,c) -> (k>>1)*288 + 2c + (k&1), so both the
//   B-operand (A[k0,c],A[k0+1,c]) and the A^T-operand (A[k0,m],A[k0+1,m])
//   are contiguous; pair-stride 288 (==32 mod 64 banks) keeps the two
//   half-waves on disjoint banks. LDS total ~210 KB (1 WG / WGP).
// ---------------------------------------------------------------------------
__global__ __launch_bounds__(256)
void k_lstep(const float* __restrict__ gmid, const float* __restrict__ Lin,
             float* __restrict__ slab, const unsigned int* __restrict__ maxv,
             int step) {
  extern __shared__ float smem[];
  float* Ls = smem;                    // 128 x LSTR
  float* Ap = Ls + 128 * LSTR;         // 64 pairs x PSTR
  float* Mp = Ap + 64 * PSTR;          // 64 pairs x PSTR
  const int d = blockIdx.x;
  const int tid = threadIdx.x;

  float scale = 1.f;
  if (step > 0) {
    const float mx = __uint_as_float(maxv[step - 1]);
    const int e = (mx > 0.f) ? ilogbf(mx) : 0;
    scale = exp2f((float)(-e));        // exponent-only => exact, log added later
  }

  // ---- stage L (scaled) and A_d into LDS with float4 global loads ---------
  const float* Acore = gmid + (size_t)step * CORE_ELEMS + (size_t)d * R_;
  for (int it = 0; it < 16; ++it) {            // 4096 float4 groups
    const int idx = it * 256 + tid;
    const int r = idx >> 5;                    // row 0..127
    const int c4 = (idx & 31) * 4;             // col group
    float4 lv = *(const float4*)(Lin + r * 128 + c4);
    lv.x *= scale; lv.y *= scale; lv.z *= scale; lv.w *= scale;
    *(float4*)(Ls + r * LSTR + c4) = lv;       // b128 store (LSTR%4==0)
    const float4 av = *(const float4*)(Acore + (size_t)r * (D_ * R_) + c4);
    float* ap = Ap + (r >> 1) * PSTR + (r & 1) + 2 * c4;  // pair-interleave
    ap[0] = av.x; ap[2] = av.y; ap[4] = av.z; ap[6] = av.w;
  }
  __syncthreads();

  const int wv   = tid >> 5;              // 8 waves: one 16-row strip each
  const int lane = tid & 31;
  const int half = lane >> 4;             // half-wave: K {0,1} vs {2,3}
  const int mrow = wv * 16 + (lane & 15); // WMMA A-operand: M = lane%16
  const int ncol = lane & 15;             // B/C/D operand: N = lane%16
  const int rbase = wv * 16 + 8 * half;   // C/D: row = rbase + vgpr

  // ---- M = L @ A : wave computes rows [wv*16,+16) x all 128 cols ----------
  v8f acc[8];
#pragma unroll
  for (int s = 0; s < 8; ++s) acc[s] = (v8f){0, 0, 0, 0, 0, 0, 0, 0};
  for (int kk = 0; kk < 32; ++kk) {
    const int k0 = kk * 4 + half * 2;          // even
    const int p0 = kk * 2 + half;              // K-pair index
    const v2f a = *(const v2f*)(Ls + mrow * LSTR + k0);
#pragma unroll
    for (int s = 0; s < 8; ++s) {
      const v2f b = *(const v2f*)(Ap + p0 * PSTR + 2 * (s * 16 + ncol));
      acc[s] = __builtin_amdgcn_wmma_f32_16x16x4_f32(false, a, false, b,
                                                     (short)0, acc[s],
                                                     false, false);
    }
  }
#pragma unroll
  for (int s = 0; s < 8; ++s) {                // store as K-pairs (v even)
#pragma unroll
    for (int v = 0; v < 8; v += 2) {
      v2f t; t.x = acc[s][v]; t.y = acc[s][v + 1];
      *(v2f*)(Mp + ((rbase + v) >> 1) * PSTR + 2 * (s * 16 + ncol)) = t;
    }
  }
  __syncthreads();

  // ---- P = A^T @ M : rows [wv*16,+16) x all 128 cols ----------------------
#pragma unroll
  for (int s = 0; s < 8; ++s) acc[s] = (v8f){0, 0, 0, 0, 0, 0, 0, 0};
  for (int kk = 0; kk < 32; ++kk) {
    const int p0 = kk * 2 + half;
    const v2f a = *(const v2f*)(Ap + p0 * PSTR + 2 * mrow);   // A^T operand
#pragma unroll
    for (int s = 0; s < 8; ++s) {
      const v2f b = *(const v2f*)(Mp + p0 * PSTR + 2 * (s * 16 + ncol));
      acc[s] = __builtin_amdgcn_wmma_f32_16x16x4_f32(false, a, false, b,
                                                     (short)0, acc[s],
                                                     false, false);
    }
  }
  float* out = slab + (size_t)d * (R_ * R_);
#pragma unroll
  for (int s = 0; s < 8; ++s) {
#pragma unroll
    for (int v = 0; v < 8; ++v)
      out[(rbase + v) * 128 + s * 16 + ncol] = acc[s][v];
  }
}

// ---------------------------------------------------------------------------
// L_next = sum_d slab[d]; record max|L_next| (exact max => deterministic).
// ---------------------------------------------------------------------------
__global__ __launch_bounds__(256)
void k_lreduce(const float* __restrict__ slab, float* __restrict__ Lout,
               unsigned int* __restrict__ maxslot) {
  __shared__ float bmax[256];
  const int e = blockIdx.x * 256 + threadIdx.x;  // 64 blocks -> 16384
  float s = 0.f;
  for (int d = 0; d < D_; ++d) s += slab[(size_t)d * (R_ * R_) + e];
  Lout[e] = s;
  bmax[threadIdx.x] = fabsf(s);
  __syncthreads();
  for (int o = 128; o > 0; o >>= 1) {
    if (threadIdx.x < o) bmax[threadIdx.x] = fmaxf(bmax[threadIdx.x], bmax[threadIdx.x + o]);
    __syncthreads();
  }
  if (threadIdx.x == 0) atomicMax(maxslot, __float_as_uint(bmax[0]));
}

// ---------------------------------------------------------------------------
// partialz[d] = v_d^T L v_d ,  v_d[p] = g_last[p, d, 0]
// ---------------------------------------------------------------------------
__global__ __launch_bounds__(128)
void k_z(const float* __restrict__ L, const float* __restrict__ glast,
         float* __restrict__ partialz) {
  __shared__ float vs[R_], red[R_];
  const int d = blockIdx.x, p = threadIdx.x;
  vs[p] = glast[p * D_ + d];
  __syncthreads();
  float w = 0.f;
  for (int q = 0; q < R_; ++q) w = fmaf(L[p * R_ + q], vs[q], w);
  red[p] = vs[p] * w;
  __syncthreads();
  if (p == 0) {
    float s = 0.f;
    for (int i = 0; i < R_; ++i) s += red[i];
    partialz[d] = s;
  }
}

// ---------------------------------------------------------------------------
// loss = log z_raw + sum(e_i)*ln2 - (2/B) * sum_b log|psi_raw_b|
// (all normalization gammas cancel algebraically into these logs)
// ---------------------------------------------------------------------------
__global__ void k_final(const float* __restrict__ psilog,
                        const float* __restrict__ partialz,
                        const unsigned int* __restrict__ maxv,
                        float* __restrict__ out) {
  float z = 0.f;
  for (int d = 0; d < D_; ++d) z += partialz[d];
  float esum = 0.f;
  for (int i = 0; i < NMID - 1; ++i) {        // scales applied in steps 1..61
    const float mx = __uint_as_float(maxv[i]);
    esum += (float)((mx > 0.f) ? ilogbf(mx) : 0);
  }
  float ps = 0.f;
  for (int b = 0; b < B_; ++b) ps += psilog[b];
  out[0] = logf(fabsf(z)) + esum * LN2 - 2.f * (ps / (float)B_);
}

// ---------------------------------------------------------------------------
extern "C" void kernel_launch(void* const* d_in, const int* in_sizes, int n_in,
                              void* d_out, int out_size, void* d_ws, size_t ws_size,
                              hipStream_t stream) {
  const int*   x     = (const int*)d_in[0];
  const float* g0    = (const float*)d_in[1];
  const float* gmid  = (const float*)d_in[2];
  const float* glast = (const float*)d_in[3];

  // workspace layout (floats): L ping-pong | slab(8MB) | maxv | partialz | psilog
  float* ws = (float*)d_ws;
  float* Lbuf0 = ws;
  float* Lbuf1 = ws + 16384;
  float* slab  = ws + 32768;
  unsigned int* maxv = (unsigned int*)(ws + 32768 + 128 * 16384);
  float* partialz = (float*)(maxv + 64);
  float* psilog   = partialz + 128;

  const size_t lds_bytes =
      (size_t)(128 * LSTR + 2 * 64 * PSTR) * sizeof(float);  // ~210 KB
  (void)hipFuncSetAttribute((const void*)k_lstep,
                            hipFuncAttributeMaxDynamicSharedMemorySize,
                            (int)lds_bytes);

  k_init<<<1, 64, 0, stream>>>(maxv);
  k_psi<<<B_, 128, 0, stream>>>(x, g0, gmid, glast, psilog);
  k_linit<<<64, 256, 0, stream>>>(g0, Lbuf0);

  float* Lc = Lbuf0;
  float* Ln = Lbuf1;
  for (int step = 0; step < NMID; ++step) {
    k_lstep<<<128, 256, lds_bytes, stream>>>(gmid, Lc, slab, maxv, step);
    k_lreduce<<<64, 256, 0, stream>>>(slab, Ln, maxv + step);
    float* t = Lc; Lc = Ln; Ln = t;
  }
  k_z<<<128, 128, 0, stream>>>(Lc, glast, partialz);
  k_final<<<1, 1, 0, stream>>>(psilog, partialz, maxv, (float*)d_out);
}